// FusionWavelets_1288490189048
// MI455X (gfx1250) — compile-verified
//
#include <hip/hip_runtime.h>

// Complex Daubechies wavelet decomposition, 6 levels, f32.
// Conv-as-matmul via V_WMMA_F32_16X16X4_F32 (wave32):
//   D[16 outputs x 4 comps(lr,li,hr,hi)] = A[16 x 12 window reals] * B[12 x 4 coeffs]
// computed as 3 chained K=4 WMMAs. B is a constant coefficient fragment.
//
// d_ws usage: ping-pong LL buffers, bufA = 24*2*512*512 floats (50.3MB),
// bufB = 24*2*256*256 floats (12.6MB) right after it. Requires ws_size >= ~63MB.

typedef __attribute__((ext_vector_type(2))) float v2f;
typedef __attribute__((ext_vector_type(8))) float v8f;

#define TILE    16
#define INROWS  38   // 2*TILE + 6 (6-tap window, stride 2)
#define MIDROWS 32   // 2*TILE  (rows 0..15 lowpass, 16..31 highpass)
#define PITCH   39   // LDS pitch: 2*39 row stride is conflict-free over 16 lanes

__global__ __launch_bounds__(256) void wavelet_level_kernel(
    const float* __restrict__ srcRe, const float* __restrict__ srcIm,
    int srcRowPitch, long long srcBcPitch,
    float* __restrict__ out,        // d_out base: [24][2][1024][1024]
    float* __restrict__ ll,         // compact LL out [24][2][h/2][w/2], or null
    const float* __restrict__ lo,   // [2][6] row-major: kr, ki
    const float* __restrict__ hi,   // [2][6]
    int h, int w)
{
  __shared__ float lds[(2 * INROWS + 2 * MIDROWS) * PITCH];
  float* inRe  = lds;
  float* inIm  = lds + INROWS * PITCH;
  float* midRe = lds + 2 * INROWS * PITCH;
  float* midIm = midRe + MIDROWS * PITCH;

  const int bc   = blockIdx.z;
  const int yo0  = blockIdx.y * TILE;
  const int xo0  = blockIdx.x * TILE;
  const int tid  = threadIdx.x;
  const int lane = tid & 31;
  const int wv   = tid >> 5;          // 8 waves per workgroup
  const int n    = lane & 15;         // A: row M; B/D: column N
  const int half = lane >> 4;         // K half-group: vgpr0 holds K, K+2

  // ---------------- B fragments: 12x4 coefficient matrix (cols 4..15 zero) ----
  // col 0 (lr): re*kr_lo  - im*ki_lo
  // col 1 (li): re*ki_lo  + im*kr_lo
  // col 2 (hr): re*kr_hi  - im*ki_hi
  // col 3 (hi): re*ki_hi  + im*kr_hi
  const float scale = 0.70710678118654752440f; // 2^-0.5 applied once to filters
  v2f bfrag[3];
#pragma unroll
  for (int c = 0; c < 3; ++c) {
#pragma unroll
    for (int j = 0; j < 2; ++j) {
      int K = 4 * c + j + 2 * half;        // B layout: vgpr j, lanes half -> K
      int t = (K < 6) ? K : K - 6;
      bool isIm = (K >= 6);
      float v = 0.0f;
      if      (n == 0) v = isIm ? -lo[6 + t] : lo[t];
      else if (n == 1) v = isIm ?  lo[t]     : lo[6 + t];
      else if (n == 2) v = isIm ? -hi[6 + t] : hi[t];
      else if (n == 3) v = isIm ?  hi[t]     : hi[6 + t];
      v *= scale;
      if (j == 0) bfrag[c].x = v; else bfrag[c].y = v;
    }
  }

  // ---------------- Phase 1: load 38x38 wrapped input tile (re, im) ----------
  {
    const int hm = h - 1, wm = w - 1;            // h, w are powers of two
    const int baseRow = 2 * yo0 - 3;
    const int baseCol = 2 * xo0 - 3;
    const float* sre = srcRe + (long long)bc * srcBcPitch;
    if (srcIm) {
      const float* sim = srcIm + (long long)bc * srcBcPitch;
      for (int idx = tid; idx < INROWS * INROWS; idx += 256) {
        int r = idx / INROWS, cc = idx - r * INROWS;
        int gr = (baseRow + r) & hm, gc = (baseCol + cc) & wm;
        long long off = (long long)gr * srcRowPitch + gc;
        inRe[r * PITCH + cc] = sre[off];
        inIm[r * PITCH + cc] = sim[off];
      }
    } else {
      for (int idx = tid; idx < INROWS * INROWS; idx += 256) {
        int r = idx / INROWS, cc = idx - r * INROWS;
        int gr = (baseRow + r) & hm, gc = (baseCol + cc) & wm;
        inRe[r * PITCH + cc] = sre[(long long)gr * srcRowPitch + gc];
        inIm[r * PITCH + cc] = 0.0f;
      }
    }
  }
  __syncthreads();

  // ---------------- Phase 2: column pass (conv along y), 38 WMMA triples -----
  for (int col = wv; col < INROWS; col += 8) {
    v8f acc = {0.f, 0.f, 0.f, 0.f, 0.f, 0.f, 0.f, 0.f};
#pragma unroll
    for (int c = 0; c < 3; ++c) {
      int K0 = 4 * c + 2 * half;
      v2f a;
      {
        int K = K0;
        int t = (K < 6) ? K : K - 6;
        const float* pl = (K < 6) ? inRe : inIm;
        a.x = pl[(2 * n + t) * PITCH + col];   // input row 2*M + t (M = n)
      }
      {
        int K = K0 + 1;
        int t = (K < 6) ? K : K - 6;
        const float* pl = (K < 6) ? inRe : inIm;
        a.y = pl[(2 * n + t) * PITCH + col];
      }
      acc = __builtin_amdgcn_wmma_f32_16x16x4_f32(
          false, a, false, bfrag[c], (short)0, acc, false, false);
    }
    if (n < 4) {                               // D cols 0..3 are the 4 comps
      float* plane = (n & 1) ? midIm : midRe;  // li/hi -> im, lr/hr -> re
      int rowoff = (n >= 2) ? TILE : 0;        // hr/hi -> highpass rows
#pragma unroll
      for (int r = 0; r < 8; ++r)
        plane[(rowoff + r + 8 * half) * PITCH + col] = acc[r];
    }
  }
  __syncthreads();

  // ---------------- Phase 3: row pass (conv along x), 32 WMMA triples --------
  const long long HW = 1024ll * 1024ll;
  const int h2 = h >> 1, w2 = w >> 1;
  for (int row = wv; row < MIDROWS; row += 8) {
    v8f acc = {0.f, 0.f, 0.f, 0.f, 0.f, 0.f, 0.f, 0.f};
#pragma unroll
    for (int c = 0; c < 3; ++c) {
      int K0 = 4 * c + 2 * half;
      v2f a;
      {
        int K = K0;
        int t = (K < 6) ? K : K - 6;
        const float* pl = (K < 6) ? midRe : midIm;
        a.x = pl[row * PITCH + 2 * n + t];     // window col 2*M + t (M = n)
      }
      {
        int K = K0 + 1;
        int t = (K < 6) ? K : K - 6;
        const float* pl = (K < 6) ? midRe : midIm;
        a.y = pl[row * PITCH + 2 * n + t];
      }
      acc = __builtin_amdgcn_wmma_f32_16x16x4_f32(
          false, a, false, bfrag[c], (short)0, acc, false, false);
    }
    if (n < 4) {
      int grow = (row < TILE) ? (yo0 + row) : (h2 + yo0 + (row - TILE));
      int p = n & 1;                            // real / imag plane
      float* pb = out + ((long long)bc * 2 + p) * HW + (long long)grow * 1024;
      int cbase = xo0 + ((n >= 2) ? w2 : 0);    // col low / high subband
      float* lb = nullptr;
      if (ll && n < 2 && row < TILE)            // LL quadrant -> next level src
        lb = ll + ((long long)bc * 2 + p) * ((long long)h2 * w2)
                + (long long)(yo0 + row) * w2 + xo0;
#pragma unroll
      for (int r = 0; r < 8; ++r) {
        int Mo = r + 8 * half;
        float v = acc[r];
        pb[cbase + Mo] = v;
        if (lb) lb[Mo] = v;
      }
    }
  }
}

extern "C" void kernel_launch(void* const* d_in, const int* in_sizes, int n_in,
                              void* d_out, int out_size, void* d_ws, size_t ws_size,
                              hipStream_t stream) {
  (void)in_sizes; (void)n_in; (void)out_size; (void)ws_size;
  const float* images = (const float*)d_in[0];
  const float* lo     = (const float*)d_in[1];
  const float* hi     = (const float*)d_in[2];
  float* out = (float*)d_out;

  float* bufA = (float*)d_ws;                          // 24*2*512*512 floats
  float* bufB = bufA + (size_t)24 * 2 * 512 * 512;     // 24*2*256*256 floats

  for (int lvl = 0; lvl < 6; ++lvl) {
    int h = 1024 >> lvl, w = 1024 >> lvl;
    const float *sre, *sim;
    long long bcPitch;
    int rowPitch;
    if (lvl == 0) {
      sre = images; sim = nullptr;
      bcPitch = 1024ll * 1024ll; rowPitch = 1024;
    } else {
      float* src = (lvl & 1) ? bufA : bufB;            // lvl1,3,5 read A; 2,4 read B
      sre = src; sim = src + (size_t)h * w;            // planes interleaved per bc
      bcPitch = 2ll * h * w; rowPitch = w;
    }
    float* llbuf = (lvl == 5) ? nullptr : ((lvl & 1) ? bufB : bufA);
    dim3 grid(w / 32, h / 32, 24);
    wavelet_level_kernel<<<grid, 256, 0, stream>>>(
        sre, sim, rowPitch, bcPitch, out, llbuf, lo, hi, h, w);
  }
}